// CRF_64802466562823
// MI455X (gfx1250) — compile-verified
//
#include <hip/hip_runtime.h>
#include <hip/hip_bf16.h>

typedef __attribute__((ext_vector_type(2))) float v2f;
typedef __attribute__((ext_vector_type(4))) float v4f;
typedef __attribute__((ext_vector_type(8))) float v8f;

#define B_TOT     256
#define T_LEN     1024
#define N_TAG     128
#define BCOL      16      // batch columns per workgroup (WMMA N)
#define START_IDX 1
#define END_IDX   2
#define NORM_MASK 3       // renormalize every 4 steps (fp32 range: growth <= ~5e4/step)

// One workgroup = 16 batch columns, 8 waves (wave w owns rows [16w,16w+16)).
// Exp-domain CRF forward with ping-pong state buffers:
//   raw = exp(tr) @ v      (32x f32 WMMA, two independent accumulation chains)
//   vnew = raw * exp(unary_t); masked columns copy old state.
// Only every 4th step does the per-column max/renormalize + log-scale update,
// so 3 of 4 steps need a single barrier and no reduction.
__global__ __launch_bounds__(256) void crf_fwd_wmma_f32(
    const float* __restrict__ unary,   // [B,T,N]
    const float* __restrict__ trans,   // [1,N,N]  (tr[i][j] = cur i, prev j)
    const int*   __restrict__ lengths, // [B]
    float* __restrict__ out)           // [B]
{
  __shared__ float vbuf[2][N_TAG][BCOL]; // ping-pong v[j][b] (2 x 8 KiB, conflict-free)
  __shared__ float red[16][BCOL];        // per (wave,half) column max
  __shared__ float s_lds[BCOL];          // per-column log scale

  const int tid  = threadIdx.x;
  const int wave = tid >> 5;             // 0..7  -> i-tile
  const int lane = tid & 31;
  const int col  = lane & 15;            // WMMA N index = batch column
  const int half = lane >> 4;            // 0/1
  const int bg   = blockIdx.x * BCOL + col;

  // ---- Preload A fragments once: A = exp(tr) tile [16 x 128], K chunks of 4.
  // 32-bit 16x4 A layout: lane<16 holds K=0,1 ; lane>=16 holds K=2,3 (M = lane&15).
  const int i_row = wave * 16 + col;
  v2f a[32];
  #pragma unroll
  for (int k = 0; k < 32; ++k) {
    const int j0 = k * 4 + half * 2;
    a[k].x = __expf(trans[i_row * N_TAG + j0 + 0]);
    a[k].y = __expf(trans[i_row * N_TAG + j0 + 1]);
  }

  // ---- Init v (exp of initial alphas: 1 at START, 0 elsewhere) and s.
  for (int idx = tid; idx < N_TAG * BCOL; idx += blockDim.x) {
    const int j = idx / BCOL, b = idx % BCOL;
    vbuf[0][j][b] = (j == START_IDX) ? 1.0f : 0.0f;
  }
  if (tid < BCOL) s_lds[tid] = 0.0f;

  // Block-uniform early exit: steps beyond max length in this tile are identity.
  int maxlen = 0;
  #pragma unroll
  for (int b = 0; b < BCOL; ++b) {
    const int L = lengths[blockIdx.x * BCOL + b];
    maxlen = (L > maxlen) ? L : maxlen;
  }
  const int mylen = lengths[bg];

  __syncthreads();

  // C/D 16x16 f32 layout: VGPR r -> row (wave*16 + half*8 + r), column = col.
  const int row0 = wave * 16 + half * 8;

  float (*vp)[BCOL] = vbuf[0];   // read buffer
  float (*vq)[BCOL] = vbuf[1];   // write buffer

  for (int t = 0; t < maxlen; ++t) {
    // Global unary loads first: latency hides under the LDS reads + GEMM.
    const float* up = unary + ((size_t)bg * T_LEN + t) * (size_t)N_TAG + row0;
    const v4f u0 = *(const v4f*)(up);
    const v4f u1 = *(const v4f*)(up + 4);

    // B fragments (all j for this column, split by half-lane K pairs).
    v2f bfr[32];
    #pragma unroll
    for (int k = 0; k < 32; ++k) {
      const int j0 = k * 4 + half * 2;
      bfr[k].x = vp[j0 + 0][col];
      bfr[k].y = vp[j0 + 1][col];
    }
    // Old state for this lane's output cells (copied through for masked columns).
    float old8[8];
    #pragma unroll
    for (int r = 0; r < 8; ++r) old8[r] = vp[row0 + r][col];

    // raw[i,b] = sum_j exp(tr[i,j]) * v[j,b] : two independent WMMA chains.
    v8f c0 = {}, c1 = {};
    #pragma unroll
    for (int k = 0; k < 16; ++k) {
      c0 = __builtin_amdgcn_wmma_f32_16x16x4_f32(false, a[2 * k], false, bfr[2 * k],
                                                 (short)0, c0, false, false);
      c1 = __builtin_amdgcn_wmma_f32_16x16x4_f32(false, a[2 * k + 1], false, bfr[2 * k + 1],
                                                 (short)0, c1, false, false);
    }
    const v8f c = c0 + c1;

    float tn[8];
    tn[0] = c[0] * __expf(u0.x);  tn[1] = c[1] * __expf(u0.y);
    tn[2] = c[2] * __expf(u0.z);  tn[3] = c[3] * __expf(u0.w);
    tn[4] = c[4] * __expf(u1.x);  tn[5] = c[5] * __expf(u1.y);
    tn[6] = c[6] * __expf(u1.z);  tn[7] = c[7] * __expf(u1.w);

    const bool active = (t < mylen);   // length mask (per column)

    if ((t & NORM_MASK) == NORM_MASK) {
      // ---- Renormalization step: per-column max across 8 waves x 2 halves.
      float pm = tn[0];
      #pragma unroll
      for (int r = 1; r < 8; ++r) pm = fmaxf(pm, tn[r]);
      red[wave * 2 + half][col] = pm;
      __syncthreads();                 // red visible; also fences vp reads

      float m = red[0][col];
      #pragma unroll
      for (int r = 1; r < 16; ++r) m = fmaxf(m, red[r][col]);
      m = fmaxf(m, 1e-30f);
      const float inv = 1.0f / m;

      #pragma unroll
      for (int r = 0; r < 8; ++r)
        vq[row0 + r][col] = active ? tn[r] * inv : old8[r];
      if (tid < BCOL && active) s_lds[col] += __logf(m);  // tid<16 -> col==tid
    } else {
      // ---- Plain step: no reduction, single barrier (ping-pong removes hazard).
      #pragma unroll
      for (int r = 0; r < 8; ++r)
        vq[row0 + r][col] = active ? tn[r] : old8[r];
    }
    __syncthreads();                   // writes visible before next step's reads

    float (*tmp)[BCOL] = vp; vp = vq; vq = tmp;
  }

  // out[b] = s[b] + log( sum_j v[j,b] * exp(tr[END, j]) )  (v may be un-normalized;
  // exact either way since s tracks only applied scales)
  if (tid < BCOL) {
    float ssum = 0.0f;
    for (int j = 0; j < N_TAG; ++j)
      ssum += vp[j][tid] * __expf(trans[END_IDX * N_TAG + j]);
    out[blockIdx.x * BCOL + tid] = s_lds[tid] + __logf(ssum);
  }
}

extern "C" void kernel_launch(void* const* d_in, const int* in_sizes, int n_in,
                              void* d_out, int out_size, void* d_ws, size_t ws_size,
                              hipStream_t stream) {
  (void)in_sizes; (void)n_in; (void)out_size; (void)d_ws; (void)ws_size;
  const float* unary   = (const float*)d_in[0];
  const float* trans   = (const float*)d_in[1];
  const int*   lengths = (const int*)d_in[2];
  float*       out     = (float*)d_out;

  dim3 grid(B_TOT / BCOL);   // 16 workgroups, one per 16-batch tile
  dim3 block(256);           // 8 wave32s, wave w owns state rows [16w, 16w+16)
  crf_fwd_wmma_f32<<<grid, block, 0, stream>>>(unary, trans, lengths, out);
}